// DFF_65266323030044
// MI455X (gfx1250) — compile-verified
//
#include <hip/hip_runtime.h>
#include <math.h>

typedef __attribute__((ext_vector_type(2))) float v2f;
typedef __attribute__((ext_vector_type(8))) float v8f;

#define B_   32
#define CF_  768
#define HW_  784
#define M_   32
#define NH_  1024
#define JTOT (M_ * CF_)          // 24576
#define BETA 0.05f

// ---------------------------------------------------------------------------
// f32 WMMA 16x16x4 wrapper (exact f32 math; emits v_wmma_f32_16x16x4_f32)
// ---------------------------------------------------------------------------
__device__ __forceinline__ v8f wmma_f32_4(v2f a, v2f b, v8f c) {
    return __builtin_amdgcn_wmma_f32_16x16x4_f32(
        /*neg_a=*/false, a, /*neg_b=*/false, b,
        /*c_mod=*/(short)0, c, /*reuse_a=*/false, /*reuse_b=*/false);
}

__device__ __forceinline__ float sigf(float x) { return 1.0f / (1.0f + __expf(-x)); }

// ---------------------------------------------------------------------------
// Kernel Z: zero [acc(64) | H0 | H1 | Cstate] region of workspace
// ---------------------------------------------------------------------------
__global__ void k_zero(float* ws, int n) {
    int i = blockIdx.x * blockDim.x + threadIdx.x;
    if (i < n) ws[i] = 0.0f;
}

// ---------------------------------------------------------------------------
// Kernel 1: BAP einsum  V[b,m,c] = (1/784) * sum_hw F[b,c,hw] * A[b,m,hw]
// grid: 32*48 blocks, 64 threads (2 waves; wave = M-tile).
// Per-wave tile: 16(M) x 16(N=c) over K=784 via f32 WMMA.
// ---------------------------------------------------------------------------
__global__ void k_bap(const float* __restrict__ F, const float* __restrict__ A,
                      float* __restrict__ V) {
    const int b  = blockIdx.x / (CF_ / 16);
    const int ct = blockIdx.x % (CF_ / 16);
    const int wave = threadIdx.x >> 5;
    const int lane = threadIdx.x & 31;
    const int m0 = wave * 16;

    const int lr = lane & 15;          // row-in-tile (A) / col-in-tile (B)
    const int kq = (lane >> 4) * 2;    // K sub-offset: 0 or 2

    const float* __restrict__ Arow = A + ((size_t)b * M_  + (m0 + lr)) * HW_;
    const float* __restrict__ Frow = F + ((size_t)b * CF_ + (ct * 16 + lr)) * HW_;

    v8f acc;
#pragma unroll
    for (int r = 0; r < 8; ++r) acc[r] = 0.0f;

#pragma unroll 4
    for (int k = 0; k < HW_; k += 4) {
        __builtin_prefetch(Frow + k + 128, 0, 0);   // stream F ahead (HBM-bound stage)
        v2f av; av[0] = Arow[k + kq]; av[1] = Arow[k + kq + 1];
        v2f bv; bv[0] = Frow[k + kq]; bv[1] = Frow[k + kq + 1];
        acc = wmma_f32_4(av, bv, acc);
    }

    const float inv = 1.0f / (float)HW_;
    const int nIdx  = ct * 16 + lr;
    const int mBase = m0 + ((lane >> 4) ? 8 : 0);
#pragma unroll
    for (int r = 0; r < 8; ++r)
        V[((size_t)b * M_ + (mBase + r)) * CF_ + nIdx] = acc[r] * inv;
}

// ---------------------------------------------------------------------------
// Kernel 2: center update + rcloss partial reduction
// 96 blocks x 256 threads; one j per thread.
// ---------------------------------------------------------------------------
__global__ void k_center(const float* __restrict__ V, const float* __restrict__ Cin,
                         float* __restrict__ Cnew_out, float* __restrict__ acc) {
    const int j = blockIdx.x * blockDim.x + threadIdx.x;   // j < 24576
    float s = 0.0f;
#pragma unroll 8
    for (int b = 0; b < B_; ++b) s += V[(size_t)b * JTOT + j];
    const float c    = Cin[j];
    const float cnew = c + BETA * (s - (float)B_ * c) / (float)B_;
    Cnew_out[j] = cnew;

    float e = 0.0f;
#pragma unroll 8
    for (int b = 0; b < B_; ++b) {
        float d = V[(size_t)b * JTOT + j] - cnew;
        e += d * d;
    }
    __shared__ float red[256];
    red[threadIdx.x] = e;
    __syncthreads();
    for (int off = 128; off > 0; off >>= 1) {
        if (threadIdx.x < off) red[threadIdx.x] += red[threadIdx.x + off];
        __syncthreads();
    }
    if (threadIdx.x == 0) atomicAdd(acc, red[0]);
}

__global__ void k_rcloss_final(const float* __restrict__ acc, float* __restrict__ out) {
    out[0] = acc[0] / ((float)B_ * (float)JTOT);
}

// ---------------------------------------------------------------------------
// Kernel 3: one LSTM step. grid = 64 blocks (N-tiles), 256 threads (8 waves).
// wave w: gate g = w&3 (0:i 1:f 2:o 3:c), M-tile mt = w>>2.
// Gate[b,n] = X_t @ Wx_g^T + bx_g + H @ Wh_g^T + bh_g  (f32 WMMA, K=768+1024)
// ---------------------------------------------------------------------------
struct LstmArgs {
    const float* V;
    const float* wx[4]; const float* bx[4];
    const float* wh[4]; const float* bh[4];
    const float* h_in;
    float* h_out;
    float* c_state;
    float* h_final;
    int t;
    int write_final;
};

__global__ void k_lstm_step(LstmArgs a) {
    const int n0   = blockIdx.x * 16;
    const int wave = threadIdx.x >> 5;
    const int lane = threadIdx.x & 31;
    const int g    = wave & 3;
    const int mt   = wave >> 2;

    const int lr = lane & 15;
    const int kq = (lane >> 4) * 2;
    const int bRow = mt * 16 + lr;     // batch row this lane supplies for A-frags
    const int nCol = n0 + lr;          // output column this lane supplies for B-frags

    const float bias = a.bx[g][nCol] + a.bh[g][nCol];
    v8f acc;
#pragma unroll
    for (int r = 0; r < 8; ++r) acc[r] = bias;

    // ---- X_t @ Wx_g^T : K = 768 ----
    const float* __restrict__ XA = a.V + ((size_t)bRow * M_ + a.t) * CF_;
    const float* __restrict__ WX = a.wx[g] + (size_t)nCol * CF_;
#pragma unroll 4
    for (int k = 0; k < CF_; k += 4) {
        v2f av; av[0] = XA[k + kq]; av[1] = XA[k + kq + 1];
        v2f bv; bv[0] = WX[k + kq]; bv[1] = WX[k + kq + 1];
        acc = wmma_f32_4(av, bv, acc);
    }
    // ---- H @ Wh_g^T : K = 1024 ----
    const float* __restrict__ HA = a.h_in + (size_t)bRow * NH_;
    const float* __restrict__ WH = a.wh[g] + (size_t)nCol * NH_;
#pragma unroll 4
    for (int k = 0; k < NH_; k += 4) {
        v2f av; av[0] = HA[k + kq]; av[1] = HA[k + kq + 1];
        v2f bv; bv[0] = WH[k + kq]; bv[1] = WH[k + kq + 1];
        acc = wmma_f32_4(av, bv, acc);
    }

    // stage gate tiles in LDS: gates[g][b 0..31][n 0..15]
    __shared__ float gates[4][B_][16];
    const int mBase = mt * 16 + ((lane >> 4) ? 8 : 0);
#pragma unroll
    for (int r = 0; r < 8; ++r) gates[g][mBase + r][lr] = acc[r];
    __syncthreads();

    // elementwise cell/hidden update: 512 elements, 2 per thread
    for (int e = threadIdx.x; e < B_ * 16; e += 256) {
        const int bb = e >> 4;
        const int nn = e & 15;
        const float ig = sigf(gates[0][bb][nn]);
        const float fg = sigf(gates[1][bb][nn]);
        const float og = sigf(gates[2][bb][nn]);
        const float cc = tanhf(gates[3][bb][nn]);
        const size_t idx = (size_t)bb * NH_ + n0 + nn;
        const float cprev = a.c_state[idx];
        const float cnew  = fg * cprev + ig * cc;
        const float h     = og * tanhf(cnew);
        a.c_state[idx] = cnew;
        a.h_out[idx]   = h;
        if (a.write_final) a.h_final[idx] = h;
    }
}

// ---------------------------------------------------------------------------
// Host-side launcher
// ---------------------------------------------------------------------------
extern "C" void kernel_launch(void* const* d_in, const int* in_sizes, int n_in,
                              void* d_out, int out_size, void* d_ws, size_t ws_size,
                              hipStream_t stream) {
    (void)in_sizes; (void)n_in; (void)out_size; (void)ws_size;

    const float* F = (const float*)d_in[0];
    const float* A = (const float*)d_in[1];
    const float* C = (const float*)d_in[2];
    // order: Wxi,bxi,Whi,bhi, Wxf,bxf,Whf,bhf, Wxo,bxo,Who,bho, Wxc,bxc,Whc,bhc
    const float* Wx[4] = { (const float*)d_in[3],  (const float*)d_in[7],
                           (const float*)d_in[11], (const float*)d_in[15] };
    const float* Bx[4] = { (const float*)d_in[4],  (const float*)d_in[8],
                           (const float*)d_in[12], (const float*)d_in[16] };
    const float* Wh[4] = { (const float*)d_in[5],  (const float*)d_in[9],
                           (const float*)d_in[13], (const float*)d_in[17] };
    const float* Bh[4] = { (const float*)d_in[6],  (const float*)d_in[10],
                           (const float*)d_in[14], (const float*)d_in[18] };

    float* out    = (float*)d_out;
    float* out_H  = out;                 // [32,1024]
    float* out_rc = out + B_ * NH_;      // [1]
    float* out_Cn = out_rc + 1;          // [24576]

    // workspace layout (floats)
    float* ws   = (float*)d_ws;
    float* acc  = ws;                    // 1 (padded to 64)
    float* H0   = ws + 64;               // 32768
    float* H1   = H0 + B_ * NH_;         // 32768
    float* Cst  = H1 + B_ * NH_;         // 32768
    float* V    = Cst + B_ * NH_;        // 786432

    const int zeroN = 64 + 3 * B_ * NH_;
    k_zero<<<(zeroN + 255) / 256, 256, 0, stream>>>(ws, zeroN);

    // Stage 1: BAP einsum (HBM-bound, f32 WMMA)
    k_bap<<<B_ * (CF_ / 16), 64, 0, stream>>>(F, A, V);

    // Stage 2: center update + rcloss
    k_center<<<JTOT / 256, 256, 0, stream>>>(V, C, out_Cn, acc);
    k_rcloss_final<<<1, 1, 0, stream>>>(acc, out_rc);

    // Stage 3: 32 sequential LSTM steps (grid-wide sync via launches; H ping-pong)
    for (int t = 0; t < M_; ++t) {
        LstmArgs la;
        la.V = V;
        for (int g = 0; g < 4; ++g) {
            la.wx[g] = Wx[g]; la.bx[g] = Bx[g];
            la.wh[g] = Wh[g]; la.bh[g] = Bh[g];
        }
        la.h_in  = (t & 1) ? H1 : H0;
        la.h_out = (t & 1) ? H0 : H1;
        la.c_state = Cst;
        la.h_final = out_H;
        la.t = t;
        la.write_final = (t == M_ - 1) ? 1 : 0;
        k_lstm_step<<<NH_ / 16, 256, 0, stream>>>(la);
    }
}